// DPSR_55130200211669
// MI455X (gfx1250) — compile-verified
//
#include <hip/hip_runtime.h>
#include <math.h>

// ---------------------------------------------------------------------------
// DPSR on MI455X (gfx1250): FFT-as-GEMM on the fp32 WMMA path, with the
// B-tile staged into LDS by the Tensor Data Mover (tensor_load_to_lds).
//   zero -> scatter -> DFTz,DFTy,DFTx (fwd, 3ch) -> spectral solve
//        -> iDFTz,iDFTy,iDFTx (1ch, last writes real) -> gather -> normalize
// Workspace requirement: ~252 MB (see offsets in kernel_launch).
// ---------------------------------------------------------------------------

#define RB 2                     // batch
#define RR 128                   // resolution per axis
#define RV (RR * RR * RR)        // 2097152 = 2^21
#define NPTS 65536
#define SIGV 10.0f
#define PI_F 3.14159265358979323846f

typedef float v2f __attribute__((ext_vector_type(2)));
typedef float v8f __attribute__((ext_vector_type(8)));
typedef unsigned int v4u __attribute__((ext_vector_type(4)));
typedef int v4i __attribute__((ext_vector_type(4)));
typedef int v8i __attribute__((ext_vector_type(8)));

// -------------------- TDM: 2-D tile load Global -> LDS ----------------------
// Builds a Tensor DMA Descriptor (D#) per CDNA5 ISA ch.8 and issues
// TENSOR_LOAD_TO_LDS. tile0 = contiguous row length (elements, 4B each),
// tile1 = number of rows, stride0 = row stride in elements.
// pad=true inserts 2 dwords of LDS padding every 128 dwords (pitch 130).
static __device__ inline void tdm_load_2d(unsigned ldsAddr, const void* gptr,
                                          unsigned tile0, unsigned tile1,
                                          unsigned stride0, bool pad) {
    unsigned long long ga = (unsigned long long)(uintptr_t)gptr;
    v4u g0;
    g0.x = 1u;                                         // count=1, user mode
    g0.y = ldsAddr;                                    // lds_addr (bytes)
    g0.z = (unsigned)ga;                               // global_addr[31:0]
    g0.w = (unsigned)((ga >> 32) & 0x01FFFFFFull)      // global_addr[56:32]
         | (2u << 30);                                 // type = 2 ("image")
    unsigned w0 = (2u << 16);                          // data_size = 4 bytes
    if (pad) w0 |= (1u << 20)                          // pad_enable
               |  (6u << 22)                           // pad_interval: 128 dw
               |  (1u << 25);                          // pad_amount: 2 dw
    v8i g1;
    g1[0] = (int)w0;                                   // mask=0 (not in cluster)
    g1[1] = (int)((tile0 & 0xFFFFu) << 16);            // tensor_dim0 = tile0
    g1[2] = (int)((tile0 >> 16) | ((tile1 & 0xFFFFu) << 16)); // dim0 hi|dim1 lo
    g1[3] = (int)((tile1 >> 16) | (tile0 << 16));      // dim1 hi | tile_dim0
    g1[4] = (int)(tile1 & 0xFFFFu);                    // tile_dim1, tile_dim2=0
    g1[5] = (int)stride0;                              // tensor_dim0_stride lo
    g1[6] = 0;                                         // stride0 hi | stride1 lo
    g1[7] = 0;
    v4i z4 = {0, 0, 0, 0};
#if __clang_major__ >= 23
    v8i z8 = {0, 0, 0, 0, 0, 0, 0, 0};
    __builtin_amdgcn_tensor_load_to_lds(g0, g1, z4, z4, z8, 0);
#else
    __builtin_amdgcn_tensor_load_to_lds(g0, g1, z4, z4, 0);
#endif
}

// -------------------- twiddle table: Wr = cos, Wi = -sin --------------------
__global__ void dpsr_twiddle(float* __restrict__ Wr, float* __restrict__ Wi) {
    int i = blockIdx.x * blockDim.x + threadIdx.x;   // 16384 threads
    if (i >= RR * RR) return;
    int f = i >> 7, z = i & 127;
    int fz = (f * z) & 127;                          // mod 128 keeps args small
    float ang = 2.0f * PI_F * (float)fz / (float)RR;
    Wr[i] = cosf(ang);
    Wi[i] = -sinf(ang);                              // forward kernel e^{-i ang}
}

// -------------------- zero ras grid + fv accumulators -----------------------
__global__ void dpsr_zero(float* __restrict__ p, long n) {
    long i = (long)blockIdx.x * blockDim.x + threadIdx.x;
    if (i < n) p[i] = 0.0f;
}

// -------------------- trilinear scatter of normals --------------------------
// ras layout: [b][c][x][y][z], c = 0..2
__global__ void dpsr_scatter(const float* __restrict__ V,
                             const float* __restrict__ N,
                             float* __restrict__ ras) {
    int i = blockIdx.x * blockDim.x + threadIdx.x;
    if (i >= RB * NPTS) return;
    int b = i / NPTS;
    const float* v = V + (long)i * 3;
    const float* n = N + (long)i * 3;
    float px = v[0] * (float)RR, py = v[1] * (float)RR, pz = v[2] * (float)RR;
    int i0x = (int)floorf(px), i0y = (int)floorf(py), i0z = (int)floorf(pz);
    int i1x = ((int)ceilf(px)) & (RR - 1);
    int i1y = ((int)ceilf(py)) & (RR - 1);
    int i1z = ((int)ceilf(pz)) & (RR - 1);
    float wx1 = px - (float)i0x, wx0 = 1.0f - wx1;   // |p - opposite corner|
    float wy1 = py - (float)i0y, wy0 = 1.0f - wy1;
    float wz1 = pz - (float)i0z, wz0 = 1.0f - wz1;
    float nx = n[0], ny = n[1], nz = n[2];
#pragma unroll
    for (int c = 0; c < 8; ++c) {
        int cx = (c >> 2) & 1, cy = (c >> 1) & 1, cz = c & 1;
        int ix = cx ? i1x : i0x;
        int iy = cy ? i1y : i0y;
        int iz = cz ? i1z : i0z;
        float w = (cx ? wx1 : wx0) * (cy ? wy1 : wy0) * (cz ? wz1 : wz0);
        long sp = (((long)ix * RR) + iy) * RR + iz;
        long base = (long)b * 3 * RV + sp;
        atomicAdd(ras + base,           w * nx);
        atomicAdd(ras + base + RV,      w * ny);
        atomicAdd(ras + base + 2 * RV,  w * nz);
    }
}

// -------------------- DFT pass along one axis via f32 WMMA ------------------
// View data as [outer][L=128][inner]; pencil p = (outer,inner).
// Output[f,p] = scale * sum_z (Wr + i*sign*Wi)[f,z] * In[z,p].
// MODE 0: Pinner==1 (K contiguous)  -> TDM tile 16 rows x 128, LDS pitch 130.
// MODE 1: Pinner>=16 (cols contig.) -> TDM tile 128 rows x 16, LDS pitch 16.
// One block = one 16-pencil column tile; 8 waves cover the 8 frequency tiles.
// B tile staged once per block by the Tensor Data Mover.
template <bool HAVE_IM, bool OUT_IM, int MODE>
__global__ __launch_bounds__(256)
void dpsr_dft_pass(const float* __restrict__ inRe,
                   const float* __restrict__ inIm,
                   float* __restrict__ outRe,
                   float* __restrict__ outIm,
                   const float* __restrict__ Wr,
                   const float* __restrict__ Wi,
                   int Pinner, float sign, float scale) {
    constexpr int LDSN = (MODE == 0) ? 16 * 130 : 128 * 16;
    __shared__ float Bre[LDSN];
    __shared__ float Bim[HAVE_IM ? LDSN : 1];

    const int ntile = blockIdx.x;                  // column (pencil) tile
    const int mtile = threadIdx.x >> 5;            // frequency tile 0..7
    const int lane  = threadIdx.x & 31;
    const int half  = lane >> 4;                   // 0: K=k0,k0+1  1: K=k0+2,+3
    const int l16   = lane & 15;
    const int p0    = ntile * 16;

    // global float-element offset of tile element (k=0, n=0)
    long tileBase;
    if (MODE == 0) {
        tileBase = (long)p0 * RR;                  // Pinner == 1
    } else {
        const int outer = p0 / Pinner;             // 16 | Pinner -> same outer
        const int inner = p0 - outer * Pinner;
        tileBase = (long)outer * RR * Pinner + inner;
    }

    if (threadIdx.x < 32) {                        // wave 0 drives the TDM
        const unsigned t0 = (MODE == 0) ? 128u : 16u;   // row length
        const unsigned t1 = (MODE == 0) ? 16u : 128u;   // row count
        const unsigned st = (MODE == 0) ? 128u : (unsigned)Pinner;
        tdm_load_2d((unsigned)(uintptr_t)&Bre[0], inRe + tileBase,
                    t0, t1, st, MODE == 0);
        if (HAVE_IM)
            tdm_load_2d((unsigned)(uintptr_t)&Bim[0], inIm + tileBase,
                        t0, t1, st, MODE == 0);
        __builtin_amdgcn_s_wait_tensorcnt(0);
    }
    __syncthreads();

    // LDS index of element (k, n):  MODE0 -> [n][130],  MODE1 -> [k][16]
    auto lidx = [](int k, int n) -> int {
        return (MODE == 0) ? (n * 130 + k) : (k * 16 + n);
    };

    const int m = mtile * 16 + l16;                // my A row (frequency)
    v8f cre = {};
    v8f cim = {};

    for (int k0 = 0; k0 < RR; k0 += 4) {
        const int k = k0 + half * 2;
        // A fragments: twiddles (16x4 f32 layout per ISA table)
        v2f ar, ai, ain;
        ar.x  = Wr[m * RR + k];
        ar.y  = Wr[m * RR + k + 1];
        ai.x  = sign * Wi[m * RR + k];
        ai.y  = sign * Wi[m * RR + k + 1];
        ain.x = -ai.x;                 // f32 WMMA has no A-negate (NEG=CNeg only)
        ain.y = -ai.y;
        // B fragments from LDS (4x16 f32 layout)
        v2f br;
        br.x = Bre[lidx(k, l16)];
        br.y = Bre[lidx(k + 1, l16)];
        cre = __builtin_amdgcn_wmma_f32_16x16x4_f32(false, ar, false, br,
                                                    (short)0, cre, false, false);
        cim = __builtin_amdgcn_wmma_f32_16x16x4_f32(false, ai, false, br,
                                                    (short)0, cim, false, false);
        if (HAVE_IM) {
            v2f bi;
            bi.x = Bim[lidx(k, l16)];
            bi.y = Bim[lidx(k + 1, l16)];
            cre = __builtin_amdgcn_wmma_f32_16x16x4_f32(false, ain, false, bi,
                                                        (short)0, cre, false, false);
            cim = __builtin_amdgcn_wmma_f32_16x16x4_f32(false, ar, false, bi,
                                                        (short)0, cim, false, false);
        }
    }

    // C/D layout: VGPR r -> M = r + 8*half, N = l16
    const int p     = p0 + l16;
    const int outer = p / Pinner;
    const int inner = p - outer * Pinner;
#pragma unroll
    for (int r = 0; r < 8; ++r) {
        const int f = mtile * 16 + r + 8 * half;
        const long o = (long)outer * RR * Pinner + (long)f * Pinner + inner;
        outRe[o] = scale * cre[r];
        if (OUT_IM) outIm[o] = scale * cim[r];
    }
}

// -------------------- pointwise spectral solve ------------------------------
// Phi = (-i * G * sum_c omega_c * Nhat_c) / (Lap + 1e-6), DC zeroed.
__global__ void dpsr_spectral(const float* __restrict__ sRe,
                              const float* __restrict__ sIm,
                              float* __restrict__ pRe,
                              float* __restrict__ pIm) {
    long i = (long)blockIdx.x * blockDim.x + threadIdx.x;
    if (i >= (long)RB * RV) return;
    int b = (int)(i >> 21);
    int s = (int)(i & (RV - 1));
    int kx = s >> 14, ky = (s >> 7) & 127, kz = s & 127;
    float fx = (kx < 64) ? (float)kx : (float)(kx - 128);   // np.fft.fftfreq order
    float fy = (ky < 64) ? (float)ky : (float)(ky - 128);
    float fz = (kz < 64) ? (float)kz : (float)(kz - 128);
    float twopi = 2.0f * PI_F;
    float wx = twopi * fx, wy = twopi * fy, wz = twopi * fz;
    float lap = -(wx * wx + wy * wy + wz * wz);
    float d2 = fx * fx + fy * fy + fz * fz;
    float a = 2.0f * SIGV / (float)RR;
    float g = __expf(-0.5f * a * a * d2);
    long base = (long)b * 3 * RV + s;
    float re = wx * sRe[base] + wy * sRe[base + RV] + wz * sRe[base + 2 * RV];
    float im = wx * sIm[base] + wy * sIm[base + RV] + wz * sIm[base + 2 * RV];
    // -i * G * (re + i*im) = G*im - i*G*re
    float inv = 1.0f / (lap + 1e-6f);
    float pr = g * im * inv;
    float pi = -g * re * inv;
    if (s == 0) { pr = 0.0f; pi = 0.0f; }
    pRe[(long)b * RV + s] = pr;
    pIm[(long)b * RV + s] = pi;
}

// -------------------- trilinear gather + per-batch fv sum -------------------
__global__ void dpsr_gather(const float* __restrict__ V,
                            const float* __restrict__ phi,
                            float* __restrict__ fvsum) {
    int i = blockIdx.x * blockDim.x + threadIdx.x;
    if (i >= RB * NPTS) return;
    int b = i / NPTS;
    const float* v = V + (long)i * 3;
    float px = v[0] * (float)RR, py = v[1] * (float)RR, pz = v[2] * (float)RR;
    int i0x = (int)floorf(px), i0y = (int)floorf(py), i0z = (int)floorf(pz);
    int i1x = ((int)ceilf(px)) & (RR - 1);
    int i1y = ((int)ceilf(py)) & (RR - 1);
    int i1z = ((int)ceilf(pz)) & (RR - 1);
    float wx1 = px - (float)i0x, wx0 = 1.0f - wx1;
    float wy1 = py - (float)i0y, wy0 = 1.0f - wy1;
    float wz1 = pz - (float)i0z, wz0 = 1.0f - wz1;
    float fv = 0.0f;
#pragma unroll
    for (int c = 0; c < 8; ++c) {
        int cx = (c >> 2) & 1, cy = (c >> 1) & 1, cz = c & 1;
        int ix = cx ? i1x : i0x;
        int iy = cy ? i1y : i0y;
        int iz = cz ? i1z : i0z;
        float w = (cx ? wx1 : wx0) * (cy ? wy1 : wy0) * (cz ? wz1 : wz0);
        long sp = (((long)ix * RR) + iy) * RR + iz;
        fv += w * phi[(long)b * RV + sp];
    }
    atomicAdd(fvsum + b, fv);
}

// -------------------- final normalization -----------------------------------
__global__ void dpsr_final(const float* __restrict__ phi,
                           const float* __restrict__ fvsum,
                           float* __restrict__ out) {
    long i = (long)blockIdx.x * blockDim.x + threadIdx.x;
    if (i >= (long)RB * RV) return;
    int b = (int)(i >> 21);
    float offset = fvsum[b] / (float)NPTS;
    float fv0 = phi[(long)b * RV] - offset;          // phi[b,0,0,0] after offset
    out[i] = -(phi[i] - offset) / fabsf(fv0) * 0.5f;
}

// ---------------------------------------------------------------------------
extern "C" void kernel_launch(void* const* d_in, const int* in_sizes, int n_in,
                              void* d_out, int out_size, void* d_ws, size_t ws_size,
                              hipStream_t stream) {
    const float* V = (const float*)d_in[0];
    const float* N = (const float*)d_in[1];
    float* out = (float*)d_out;

    // workspace layout (floats)
    float* ws    = (float*)d_ws;
    const long TW = (long)RR * RR;                 // 16384
    const long CH = (long)RB * 3 * RV;             // 6 * 2^21
    float* Wr    = ws;
    float* Wi    = Wr + TW;
    float* ras   = Wi + TW;          // [b][c][x][y][z], 6V
    float* S1re  = ras + CH;         // 6V
    float* S1im  = S1re + CH;        // 6V
    float* S2re  = S1im + CH;        // 6V
    float* S2im  = S2re + CH;        // 6V
    float* fvsum = S2im + CH;        // RB

    // 1) twiddles
    dpsr_twiddle<<<dim3((TW + 255) / 256), dim3(256), 0, stream>>>(Wr, Wi);

    // 2) zero ras + fvsum
    dpsr_zero<<<dim3((unsigned)((CH + 255) / 256)), dim3(256), 0, stream>>>(ras, CH);
    dpsr_zero<<<dim3(1), dim3(32), 0, stream>>>(fvsum, RB);

    // 3) scatter points
    dpsr_scatter<<<dim3((RB * NPTS + 255) / 256), dim3(256), 0, stream>>>(V, N, ras);

    // 4-6) forward DFT passes over z, y, x (3 channels * 2 batches folded)
    const int fwdCols = RB * 3 * RR * RR / 16;     // 6144 blocks (8 waves each)
    // z: [outer=B*C*128*128][z][inner=1]  (r2c, K-contiguous tiles)
    dpsr_dft_pass<false, true, 0><<<dim3(fwdCols), dim3(256), 0, stream>>>(
        ras, nullptr, S1re, S1im, Wr, Wi, 1, 1.0f, 1.0f);
    // y: [outer=B*C*128][y][inner=128]
    dpsr_dft_pass<true, true, 1><<<dim3(fwdCols), dim3(256), 0, stream>>>(
        S1re, S1im, S2re, S2im, Wr, Wi, RR, 1.0f, 1.0f);
    // x: [outer=B*C][x][inner=128*128]
    dpsr_dft_pass<true, true, 1><<<dim3(fwdCols), dim3(256), 0, stream>>>(
        S2re, S2im, S1re, S1im, Wr, Wi, RR * RR, 1.0f, 1.0f);

    // 7) spectral solve: S1 (3ch spectra) -> S2 (Phi spectrum, 2V used)
    dpsr_spectral<<<dim3((unsigned)(((long)RB * RV + 255) / 256)), dim3(256), 0, stream>>>(
        S1re, S1im, S2re, S2im);

    // 8-10) inverse DFT passes (sign=-1 -> e^{+i}, scale 1/128 each)
    const int invCols = RB * RR * RR / 16;         // 2048 blocks
    const float is = 1.0f / (float)RR;
    dpsr_dft_pass<true, true, 0><<<dim3(invCols), dim3(256), 0, stream>>>(
        S2re, S2im, S1re, S1im, Wr, Wi, 1, -1.0f, is);
    dpsr_dft_pass<true, true, 1><<<dim3(invCols), dim3(256), 0, stream>>>(
        S1re, S1im, S2re, S2im, Wr, Wi, RR, -1.0f, is);
    // last pass: write real part only -> phi in S1re (2V)
    dpsr_dft_pass<true, false, 1><<<dim3(invCols), dim3(256), 0, stream>>>(
        S2re, S2im, S1re, nullptr, Wr, Wi, RR * RR, -1.0f, is);

    // 11) gather fv sums
    dpsr_gather<<<dim3((RB * NPTS + 255) / 256), dim3(256), 0, stream>>>(V, S1re, fvsum);

    // 12) normalize -> d_out
    dpsr_final<<<dim3((unsigned)(((long)RB * RV + 255) / 256)), dim3(256), 0, stream>>>(
        S1re, fvsum, out);
}